// PDN_33457795236019
// MI455X (gfx1250) — compile-verified
//
#include <hip/hip_runtime.h>
#include <hip/hip_bf16.h>

typedef __attribute__((ext_vector_type(2))) float v2f;
typedef __attribute__((ext_vector_type(8))) float v8f;

// D(16x16,f32) += A(16x4,f32) * B(4x16,f32)
// Per ISA 7.12.2: A: lane m=lane&15 holds A[m][k0],A[m][k0+1] with k0=2*(lane>>4)
//                 B: lane n=lane&15 holds B[k0][n],B[k0+1][n]
//                 D: 8 vgprs, row = r + 8*(lane>=16), col = lane&15
__device__ __forceinline__ v8f wmma4(v2f a, v2f b, v8f c) {
  return __builtin_amdgcn_wmma_f32_16x16x4_f32(false, a, false, b, (short)0, c, false, false);
}

// bilinear (align_corners=False, scale 2: 14 -> 28) tap setup
__device__ __forceinline__ void bl_setup(int h, int w,
    int& i00, int& i01, int& i10, int& i11,
    float& w00, float& w01, float& w10, float& w11) {
  float fh = h * 0.5f - 0.25f;
  float fw = w * 0.5f - 0.25f;
  int h0 = (int)floorf(fh), w0 = (int)floorf(fw);
  float ah = fh - (float)h0, aw = fw - (float)w0;
  int h0c = max(0, min(13, h0)), h1c = max(0, min(13, h0 + 1));
  int w0c = max(0, min(13, w0)), w1c = max(0, min(13, w0 + 1));
  i00 = h0c * 14 + w0c; i01 = h0c * 14 + w1c;
  i10 = h1c * 14 + w0c; i11 = h1c * 14 + w1c;
  w00 = (1.f - ah) * (1.f - aw); w01 = (1.f - ah) * aw;
  w10 = ah * (1.f - aw);         w11 = ah * aw;
}

__device__ __forceinline__ float bl_at(const float* p, int hw) {
  int h = hw / 28, w = hw % 28;
  int i00, i01, i10, i11; float w00, w01, w10, w11;
  bl_setup(h, w, i00, i01, i10, i11, w00, w01, w10, w11);
  return w00 * p[i00] + w01 * p[i01] + w10 * p[i10] + w11 * p[i11];
}

// ---------------- kernel 0: a_sq[l] = sum_c conv_w[l][c], padded rows = +1e30
__global__ void k0_asq(const float* __restrict__ conv_w, float* __restrict__ asq) {
  int lane = threadIdx.x;          // 32 threads
  int l = blockIdx.x;              // 16 blocks
  float s = 0.f;
  if (l < 9) {
    const float* r = conv_w + (size_t)l * 3072;
    for (int c = lane; c < 3072; c += 32) s += r[c];
  }
  s += __shfl_xor(s, 1); s += __shfl_xor(s, 2); s += __shfl_xor(s, 4);
  s += __shfl_xor(s, 8); s += __shfl_xor(s, 16);
  if (lane == 0) asq[l] = (l < 9) ? s : 1.0e30f;
}

// ---------------- kernel 1: attention maps (softmax over 9 landmarks)
// one wave = one batch b, one 16-wide spatial tile. 784 = 49*16, 49 = 7*7.
__global__ __launch_bounds__(224) void k1_maps(
    const float* __restrict__ l3, const float* __restrict__ l4,
    const float* __restrict__ conv_w, const float* __restrict__ asq,
    float* __restrict__ maps) {
  const int lane = threadIdx.x & 31;
  const int wid  = threadIdx.x >> 5;            // 0..6
  const int b    = blockIdx.x / 7;
  const int tile = (blockIdx.x % 7) * 7 + wid;  // 0..48
  const int n    = lane & 15;
  const int half = lane >> 4;
  const int hw   = tile * 16 + n;               // B column's spatial index
  // bilinear taps for this lane's spatial position (fixed over K loop)
  int i00, i01, i10, i11; float w00, w01, w10, w11;
  bl_setup(hw / 28, hw % 28, i00, i01, i10, i11, w00, w01, w10, w11);

  const float* l4b = l4 + (size_t)b * 2048 * 196;
  const float* l3b = l3 + (size_t)b * 1024 * 784;
  // A: landmark row m = lane&15 of conv_w (rows >= 9 are zero padding)
  const int   m       = n;
  const float cwvalid = (m < 9) ? 1.0f : 0.0f;
  const float* cwrow  = conv_w + (size_t)min(m, 8) * 3072;

  v8f acc = {};
  float bsq = 0.f;
  // phase 1: channels 0..2047 come from bilinear-upsampled l4
  #pragma unroll 2
  for (int k = 0; k < 2048; k += 4) {
    const int c0 = k + half * 2;
    v2f a; a.x = cwrow[c0] * cwvalid; a.y = cwrow[c0 + 1] * cwvalid;
    const float* p0 = l4b + (size_t)c0 * 196;
    const float* p1 = p0 + 196;
    v2f bb;
    bb.x = w00 * p0[i00] + w01 * p0[i01] + w10 * p0[i10] + w11 * p0[i11];
    bb.y = w00 * p1[i00] + w01 * p1[i01] + w10 * p1[i10] + w11 * p1[i11];
    bsq += bb.x * bb.x + bb.y * bb.y;
    acc = wmma4(a, bb, acc);
  }
  // phase 2: channels 2048..3071 come directly from l3
  #pragma unroll 2
  for (int k = 0; k < 1024; k += 4) {
    const int c0 = k + half * 2;
    v2f a; a.x = cwrow[2048 + c0] * cwvalid; a.y = cwrow[2048 + c0 + 1] * cwvalid;
    const float* p = l3b + (size_t)c0 * 784 + hw;
    v2f bb; bb.x = p[0]; bb.y = p[784];
    bsq += bb.x * bb.x + bb.y * bb.y;
    acc = wmma4(a, bb, acc);
  }
  // combine the two channel-halves of b_sq (lane n <-> lane n+16, same column)
  bsq += __shfl_xor(bsq, 16);

  // logits for the 8 rows this lane holds; padded rows get asq=+1e30 -> exp 0
  const float* asqb = asq + half * 8;
  float lg[8]; float mx = -3.0e38f;
  #pragma unroll
  for (int r = 0; r < 8; ++r) {
    lg[r] = -(bsq - 2.0f * acc[r] + asqb[r]);
    mx = fmaxf(mx, lg[r]);
  }
  mx = fmaxf(mx, __shfl_xor(mx, 16));
  float e[8]; float s = 0.f;
  #pragma unroll
  for (int r = 0; r < 8; ++r) { e[r] = __expf(lg[r] - mx); s += e[r]; }
  s += __shfl_xor(s, 16);
  const float inv = 1.0f / s;

  float* mb = maps + (size_t)b * 9 * 784 + hw;
  #pragma unroll
  for (int r = 0; r < 8; ++r) {
    int l = half * 8 + r;
    if (l < 9) mb[(size_t)l * 784] = e[r] * inv;
  }
}

// ---------------- kernel 2: all_features = x @ maps^T / 784, + mean_features
// one wave = one batch, one 16-channel tile (192 tiles). K = 784 spatial.
__global__ __launch_bounds__(256) void k2_allfeat(
    const float* __restrict__ l3, const float* __restrict__ l4,
    const float* __restrict__ maps, const float* __restrict__ modulation,
    float* __restrict__ allfeat, float* __restrict__ mf) {
  const int lane = threadIdx.x & 31;
  const int wid  = threadIdx.x >> 5;            // 0..7
  const int gw   = blockIdx.x * 8 + wid;        // 0 .. 32*192-1
  const int b    = gw / 192;
  const int cbase = (gw % 192) * 16;
  const int n    = lane & 15;
  const int half = lane >> 4;
  const int ch   = cbase + n;                   // A row: channel (tile never straddles 2048)

  // B: maps row for landmark n (9 valid, rest zeroed)
  const float mvalid = (n < 9) ? 1.0f : 0.0f;
  const float* mrow  = maps + ((size_t)b * 9 + min(n, 8)) * 784;

  v8f acc = {};
  if (__builtin_amdgcn_readfirstlane(cbase) < 2048) {
    const float* ap = l4 + ((size_t)b * 2048 + ch) * 196;
    #pragma unroll 2
    for (int k = 0; k < 784; k += 4) {
      const int k0 = k + half * 2;
      v2f bb; bb.x = mrow[k0] * mvalid; bb.y = mrow[k0 + 1] * mvalid;
      v2f a;  a.x = bl_at(ap, k0);      a.y = bl_at(ap, k0 + 1);
      acc = wmma4(a, bb, acc);
    }
  } else {
    const float* ap = l3 + ((size_t)b * 1024 + (ch - 2048)) * 784;
    #pragma unroll 2
    for (int k = 0; k < 784; k += 4) {
      const int k0 = k + half * 2;
      v2f bb; bb.x = mrow[k0] * mvalid; bb.y = mrow[k0 + 1] * mvalid;
      v2f a;  a.x = ap[k0];             a.y = ap[k0 + 1];
      acc = wmma4(a, bb, acc);
    }
  }

  const float inv784 = 1.0f / 784.0f;
  // store all_features (pre-modulation, as the reference returns it)
  #pragma unroll
  for (int r = 0; r < 8; ++r) {
    const int c = cbase + half * 8 + r;
    if (n < 9) allfeat[((size_t)b * 3072 + c) * 9 + n] = acc[r] * inv784;
  }
  // mean_features: mean over landmarks 0..7 of (af * modulation)
  #pragma unroll
  for (int r = 0; r < 8; ++r) {
    const int c = cbase + half * 8 + r;
    const float modv = modulation[(size_t)c * 9 + min(n, 8)];
    float v = (n < 8) ? acc[r] * inv784 * modv : 0.f;
    v += __shfl_xor(v, 1); v += __shfl_xor(v, 2); v += __shfl_xor(v, 4);
    if ((lane & 15) == 0)  // lane 0 -> rows 0..7, lane 16 -> rows 8..15
      mf[(size_t)b * 3072 + c] = v * 0.125f;
  }
}

// ---------------- kernel 3: attr_scores = mf @ attr_w^T + attr_b  (32x312, K=3072)
__global__ __launch_bounds__(256) void k3_attr(
    const float* __restrict__ mf, const float* __restrict__ attr_w,
    const float* __restrict__ attr_b, float* __restrict__ attr) {
  const int lane = threadIdx.x & 31;
  const int wid  = threadIdx.x >> 5;
  const int gw   = blockIdx.x * 8 + wid;        // 0..39
  const int mt   = gw / 20, nt = gw % 20;       // 2 row tiles x 20 col tiles (320 cols padded)
  const int n    = lane & 15;
  const int half = lane >> 4;
  const float* arow = mf + (size_t)(mt * 16 + n) * 3072;
  const float* brow = attr_w + (size_t)min(nt * 16 + n, 311) * 3072;
  v8f acc = {};
  #pragma unroll 4
  for (int k = 0; k < 3072; k += 4) {
    const int k0 = k + half * 2;
    v2f a;  a.x = arow[k0];  a.y = arow[k0 + 1];
    v2f bb; bb.x = brow[k0]; bb.y = brow[k0 + 1];
    acc = wmma4(a, bb, acc);
  }
  const int ocol = nt * 16 + n;
  if (ocol < 312) {
    const float bias = attr_b[ocol];
    #pragma unroll
    for (int r = 0; r < 8; ++r)
      attr[(size_t)(mt * 16 + half * 8 + r) * 312 + ocol] = acc[r] + bias;
  }
}

// ---------------- kernel 4: class_scores = attr @ class_w^T  (32x200, K=312)
__global__ __launch_bounds__(32) void k4_class(
    const float* __restrict__ attr, const float* __restrict__ class_w,
    float* __restrict__ cls) {
  const int lane = threadIdx.x & 31;
  const int mt   = blockIdx.x / 13, nt = blockIdx.x % 13;  // 2 x 13 tiles (208 cols padded)
  const int n    = lane & 15;
  const int half = lane >> 4;
  const float* arow = attr + (size_t)(mt * 16 + n) * 312;
  const float* brow = class_w + (size_t)min(nt * 16 + n, 199) * 312;
  v8f acc = {};
  #pragma unroll 4
  for (int k = 0; k < 312; k += 4) {
    const int k0 = k + half * 2;
    v2f a;  a.x = arow[k0];  a.y = arow[k0 + 1];
    v2f bb; bb.x = brow[k0]; bb.y = brow[k0 + 1];
    acc = wmma4(a, bb, acc);
  }
  const int ocol = nt * 16 + n;
  if (ocol < 200) {
    #pragma unroll
    for (int r = 0; r < 8; ++r)
      cls[(size_t)(mt * 16 + half * 8 + r) * 200 + ocol] = acc[r];
  }
}

extern "C" void kernel_launch(void* const* d_in, const int* in_sizes, int n_in,
                              void* d_out, int out_size, void* d_ws, size_t ws_size,
                              hipStream_t stream) {
  const float* l3         = (const float*)d_in[0];  // (32,1024,28,28)
  const float* l4         = (const float*)d_in[1];  // (32,2048,14,14)
  const float* conv_w     = (const float*)d_in[2];  // (9,3072)
  const float* modulation = (const float*)d_in[3];  // (1,3072,9)
  const float* attr_w     = (const float*)d_in[4];  // (312,3072)
  const float* attr_b     = (const float*)d_in[5];  // (312,)
  const float* class_w    = (const float*)d_in[6];  // (200,312)

  float* out   = (float*)d_out;
  float* attr  = out;                 // 32*312   = 9984
  float* cls   = out + 9984;          // 32*200   = 6400
  float* maps  = out + 16384;         // 32*9*784 = 225792
  float* af    = out + 242176;        // 32*3072*9

  float* wsf = (float*)d_ws;
  float* asq = wsf;                   // 16 floats (padded a_sq)
  float* mf  = wsf + 16;              // 32*3072 floats

  k0_asq   <<<16,      32,  0, stream>>>(conv_w, asq);
  k1_maps  <<<32 * 7,  224, 0, stream>>>(l3, l4, conv_w, asq, maps);
  k2_allfeat<<<32 * 24, 256, 0, stream>>>(l3, l4, maps, modulation, af, mf);
  k3_attr  <<<5,       256, 0, stream>>>(mf, attr_w, attr_b, attr);
  k4_class <<<26,      32,  0, stream>>>(attr, class_w, cls);
}